// LieSE2_37065567764675
// MI455X (gfx1250) — compile-verified
//
#include <hip/hip_runtime.h>

// SE(2)-exp elementwise kernel for MI455X (gfx1250, wave32).
// Memory-bound: 192 MB streamed once -> ~8.2 us floor at 23.3 TB/s.
// Strategy: 4 elements per thread so every global access is a 16B-aligned
// b128 op (3 NT loads + 9 NT stores per thread); hardware trans ops
// (v_sin/v_cos/v_rcp) keep VALU time far below the memory floor.

typedef __attribute__((ext_vector_type(4))) float v4f;

#define SE2_EPS 1e-5f

struct SE2Out { float r00, r01, tx, r10, r11, ty; };

__device__ __forceinline__ SE2Out se2_elem(float x, float y, float w) {
    float t2  = w * w;
    float th  = __builtin_fabsf(w);            // sqrt(w*w) == |w|
    float s   = __sinf(th);                    // v_sin_f32 (post 1/2pi reduce)
    float c   = __cosf(th);                    // v_cos_f32
    float ivt = __builtin_amdgcn_rcpf(th + SE2_EPS);  // v_rcp_f32
    float iv2 = __builtin_amdgcn_rcpf(t2 + SE2_EPS);  // shared denom
    float A   = s * ivt;
    float Bc  = (1.0f - c) * iv2;
    float C   = (1.0f - A) * iv2;
    SE2Out o;
    o.r00 = 1.0f - Bc * t2;
    o.r01 = -A * w;
    o.r10 =  A * w;
    o.r11 = o.r00;
    float v00 = 1.0f - C * t2;
    float v01 = -Bc * w;
    float v10 =  Bc * w;
    o.tx = v00 * x + v01 * y;
    o.ty = v10 * x + v00 * y;
    return o;
}

// Main kernel: thread t handles elements 4t..4t+3.
// in4 : 3 float4 per thread  (12 floats = 4 x (x,y,w))
// out4: 9 float4 per thread  (36 floats = 4 x 3x3 matrix, row-major)
__global__ __launch_bounds__(256) void se2_vec4_kernel(const v4f* __restrict__ in4,
                                                       v4f* __restrict__ out4,
                                                       int n4) {
    int t = blockIdx.x * 256 + threadIdx.x;
    if (t >= n4) return;

    v4f a = __builtin_nontemporal_load(in4 + 3 * t + 0);  // x0 y0 w0 x1
    v4f b = __builtin_nontemporal_load(in4 + 3 * t + 1);  // y1 w1 x2 y2
    v4f c = __builtin_nontemporal_load(in4 + 3 * t + 2);  // w2 x3 y3 w3

    SE2Out e0 = se2_elem(a.x, a.y, a.z);
    SE2Out e1 = se2_elem(a.w, b.x, b.y);
    SE2Out e2 = se2_elem(b.z, b.w, c.x);
    SE2Out e3 = se2_elem(c.y, c.z, c.w);

    // Per element layout: r00 r01 tx r10 r11 ty 0 0 1  (9 floats)
    v4f o0 = {e0.r00, e0.r01, e0.tx,  e0.r10};
    v4f o1 = {e0.r11, e0.ty,  0.0f,   0.0f  };
    v4f o2 = {1.0f,   e1.r00, e1.r01, e1.tx };
    v4f o3 = {e1.r10, e1.r11, e1.ty,  0.0f  };
    v4f o4 = {0.0f,   1.0f,   e2.r00, e2.r01};
    v4f o5 = {e2.tx,  e2.r10, e2.r11, e2.ty };
    v4f o6 = {0.0f,   0.0f,   1.0f,   e3.r00};
    v4f o7 = {e3.r01, e3.tx,  e3.r10, e3.r11};
    v4f o8 = {e3.ty,  0.0f,   0.0f,   1.0f  };

    v4f* p = out4 + 9 * t;
    __builtin_nontemporal_store(o0, p + 0);
    __builtin_nontemporal_store(o1, p + 1);
    __builtin_nontemporal_store(o2, p + 2);
    __builtin_nontemporal_store(o3, p + 3);
    __builtin_nontemporal_store(o4, p + 4);
    __builtin_nontemporal_store(o5, p + 5);
    __builtin_nontemporal_store(o6, p + 6);
    __builtin_nontemporal_store(o7, p + 7);
    __builtin_nontemporal_store(o8, p + 8);
}

// Scalar tail for n % 4 != 0 (not hit for B = 4,000,000, kept for generality).
__global__ void se2_tail_kernel(const float* __restrict__ in,
                                float* __restrict__ out,
                                int start, int n) {
    int i = start + blockIdx.x * blockDim.x + threadIdx.x;
    if (i >= n) return;
    SE2Out e = se2_elem(in[3 * i + 0], in[3 * i + 1], in[3 * i + 2]);
    float* p = out + 9 * i;
    p[0] = e.r00; p[1] = e.r01; p[2] = e.tx;
    p[3] = e.r10; p[4] = e.r11; p[5] = e.ty;
    p[6] = 0.0f;  p[7] = 0.0f;  p[8] = 1.0f;
}

extern "C" void kernel_launch(void* const* d_in, const int* in_sizes, int n_in,
                              void* d_out, int out_size, void* d_ws, size_t ws_size,
                              hipStream_t stream) {
    (void)n_in; (void)out_size; (void)d_ws; (void)ws_size;
    const float* in = (const float*)d_in[0];
    float* out = (float*)d_out;

    int n   = in_sizes[0] / 3;   // number of (x, y, w) rows
    int n4  = n / 4;             // vectorized groups of 4 elements
    int rem = n - 4 * n4;

    if (n4 > 0) {
        int blocks = (n4 + 255) / 256;
        se2_vec4_kernel<<<blocks, 256, 0, stream>>>((const v4f*)in, (v4f*)out, n4);
    }
    if (rem > 0) {
        se2_tail_kernel<<<1, 256, 0, stream>>>(in, out, 4 * n4, n);
    }
}